// neighborSearchModule_25374666785469
// MI455X (gfx1250) — compile-verified
//
#include <hip/hip_runtime.h>
#include <stdint.h>

// ---- Grid constants (mirror the reference) ----
#define SUPPORT_F 0.01f
#define CELLS     103
#define NUM_CELLS (CELLS * CELLS)      // 10609
#define MPC       40                   // MAX_PER_CELL
#define KSLOTS    (9 * MPC)            // 360
#define CPB       64                   // cells per sort block

typedef __attribute__((ext_vector_type(4))) unsigned int u32x4;
typedef __attribute__((ext_vector_type(8))) int          i32x8;
typedef __attribute__((ext_vector_type(4))) int          i32x4;
typedef __attribute__((ext_vector_type(2))) float        f32x2;

__device__ __forceinline__ int cellOf(float p) {
    // indices = ceil((p - QMIN) / SUPPORT), QMIN = -SUPPORT
    return (int)ceilf((p - (-SUPPORT_F)) / SUPPORT_F);
}

// ---- Kernel 1: init table to -1, counts to 0 ----
__global__ void ns_init(int* __restrict__ table, int* __restrict__ counts, int nTable) {
    int t = blockIdx.x * blockDim.x + threadIdx.x;
    if (t < nTable)    table[t]  = -1;
    if (t < NUM_CELLS) counts[t] = 0;
}

// ---- Kernel 2: counting-insert particles into cell table ----
__global__ void ns_build(const float* __restrict__ x, int n,
                         int* __restrict__ table, int* __restrict__ counts) {
    int j = blockIdx.x * blockDim.x + threadIdx.x;
    if (j >= n) return;
    int cx = cellOf(x[2 * j]);
    int cy = cellOf(x[2 * j + 1]);
    if (cx < 0 || cx >= CELLS || cy < 0 || cy >= CELLS) return;
    int lin  = cx + CELLS * cy;
    int rank = atomicAdd(&counts[lin], 1);
    if (rank < MPC) table[lin * MPC + rank] = j;
}

// ---- Kernel 3: per-cell ascending sort (restores stable-argsort slot order).
//      Rows are DMA-staged into LDS via the Tensor Data Mover. ----
__global__ void ns_sort(int* __restrict__ table, const int* __restrict__ counts) {
    __shared__ int rows[CPB * MPC];            // 10240 bytes
    const int cellBase  = blockIdx.x * CPB;
    const int tileElems = CPB * MPC;           // 2560 dwords per tile

    if (threadIdx.x < 32) {                    // wave 0 issues the TDM op
        unsigned long long gaddr =
            (unsigned long long)(uintptr_t)(table + (size_t)cellBase * MPC);
        unsigned int ldsOff = (unsigned int)(uintptr_t)(&rows[0]);

        u32x4 g0;
        g0.x = 1u;                                               // count=1, user D#
        g0.y = ldsOff;                                           // lds_addr (bytes)
        g0.z = (unsigned int)(gaddr & 0xffffffffu);              // global_addr lo
        g0.w = (unsigned int)((gaddr >> 32) & 0x1ffffffu)        // global_addr hi
             | (2u << 30);                                       // type = 2 ("image")

        i32x8 g1;
        g1[0] = (2 << 16);                                       // data_size = 4 bytes
        g1[1] = (int)((tileElems & 0xffff) << 16);               // tensor_dim0[15:0]
        g1[2] = ((tileElems >> 16) & 0xffff) | (1 << 16);        // dim0 hi | tensor_dim1=1
        g1[3] = (int)((tileElems & 0xffff) << 16);               // tile_dim0 = 2560
        g1[4] = 1;                                               // tile_dim1 = 1
        g1[5] = tileElems;                                       // tensor_dim0_stride lo
        g1[6] = 0;
        g1[7] = 0;

        i32x4 gz4 = {0, 0, 0, 0};                                // groups 2/3 unused (<=2D)
        i32x8 gz8 = {0, 0, 0, 0, 0, 0, 0, 0};
        __builtin_amdgcn_tensor_load_to_lds(g0, g1, gz4, gz4, gz8, 0);
        __builtin_amdgcn_s_wait_tensorcnt(0);
    }
    __syncthreads();

    int t = threadIdx.x;
    if (t < CPB) {
        int cell = cellBase + t;
        if (cell < NUM_CELLS) {
            int cnt = counts[cell];
            if (cnt > MPC) cnt = MPC;
            int* r = &rows[t * MPC];
            // insertion sort ascending over the valid prefix (random LDS access)
            for (int a = 1; a < cnt; ++a) {
                int v = r[a];
                int b = a - 1;
                while (b >= 0 && r[b] > v) { r[b + 1] = r[b]; --b; }
                r[b + 1] = v;
            }
            for (int a = 0; a < MPC; ++a) table[cell * MPC + a] = r[a];
        }
    }
}

// ---- Kernel 4: one thread per (query, slot); NT-streamed outputs ----
__global__ void ns_query(const float* __restrict__ x, const float* __restrict__ y,
                         const int* __restrict__ table, float* __restrict__ out,
                         long long totalMK) {
    long long gid = (long long)blockIdx.x * blockDim.x + threadIdx.x;
    if (gid >= totalMK) return;
    int i = (int)(gid / KSLOTS);
    int s = (int)(gid - (long long)i * KSLOTS);

    float yx = y[2 * i], yy = y[2 * i + 1];
    int qcx = cellOf(yx), qcy = cellOf(yy);

    int c  = s / MPC;                    // 0..8 stencil cell, dx-major like reference
    int r  = s - c * MPC;                // 0..39 slot in cell
    int cx = qcx + (c / 3) - 1;
    int cy = qcy + (c - (c / 3) * 3) - 1;

    int cand = -1;
    if (cx >= 0 && cx < CELLS && cy >= 0 && cy < CELLS)
        cand = table[(cx + CELLS * cy) * MPC + r];

    float nbrOut = -1.0f, ux = 0.0f, uy = 0.0f, rad = 0.0f, mk = 0.0f;
    if (cand >= 0) {
        float dx   = yx - x[2 * cand];
        float dy   = yy - x[2 * cand + 1];
        float dist = sqrtf(dx * dx + dy * dy);
        if (dist <= SUPPORT_F) {
            mk     = 1.0f;
            nbrOut = (float)cand;                 // < 2^24, exact in f32
            rad    = dist / SUPPORT_F;
            if (dist >= 1e-6f) {                  // eps = 1e-4 * SUPPORT
                ux = dx / dist;
                uy = dy / dist;
            }
        }
    }

    // Outputs concatenated in return order: nbr[MK], unit[MK,2], radial[MK], mask[MK].
    // 720MB stream > 192MB L2 -> non-temporal stores keep table/x hot in L2.
    __builtin_nontemporal_store(nbrOut, out + gid);
    f32x2 u; u.x = ux; u.y = uy;
    __builtin_nontemporal_store(u, (f32x2*)(out + totalMK + 2 * gid));
    __builtin_nontemporal_store(rad, out + 3 * totalMK + gid);
    __builtin_nontemporal_store(mk,  out + 4 * totalMK + gid);
}

extern "C" void kernel_launch(void* const* d_in, const int* in_sizes, int n_in,
                              void* d_out, int out_size, void* d_ws, size_t ws_size,
                              hipStream_t stream) {
    const float* x = (const float*)d_in[0];
    const float* y = (const float*)d_in[1];
    int n = in_sizes[0] / 2;
    int m = in_sizes[1] / 2;
    float* out = (float*)d_out;

    // ws layout: table first, counts after (TDM tail over-read lands in counts; safe).
    int* table  = (int*)d_ws;                   // NUM_CELLS*MPC ints
    int* counts = table + NUM_CELLS * MPC;      // NUM_CELLS ints
    const int nTable = NUM_CELLS * MPC;

    ns_init <<<(nTable + 255) / 256, 256, 0, stream>>>(table, counts, nTable);
    ns_build<<<(n + 255) / 256,      256, 0, stream>>>(x, n, table, counts);
    ns_sort <<<(NUM_CELLS + CPB - 1) / CPB, CPB, 0, stream>>>(table, counts);

    long long totalMK = (long long)m * KSLOTS;
    long long blocks  = (totalMK + 255) / 256;
    ns_query<<<(unsigned int)blocks, 256, 0, stream>>>(x, y, table, out, totalMK);
}